// GAT_23038204576109
// MI455X (gfx1250) — compile-verified
//
#include <hip/hip_runtime.h>
#include <hip/hip_bf16.h>

typedef __attribute__((ext_vector_type(2))) float v2f;
typedef __attribute__((ext_vector_type(8))) float v8f;

#define HC 128   // HEADS * C
#define CC 64    // C
#define NHEADS 2

// ---------------------------------------------------------------------------
// GEMM: H[N,128] = X[N,K] @ W[128,K]^T  via V_WMMA_F32_16X16X4_F32.
// Block = 256 threads = 8 waves; block handles a 16-row x 128-col output tile.
// Wave w owns output columns [16w, 16w+16). X tile staged in LDS.
// A layout (16x4 f32): lanes 0-15 -> M=lane, VGPR0 K=0, VGPR1 K=1;
//                      lanes 16-31 -> M=lane-16, VGPR0 K=2, VGPR1 K=3.
// B layout (4x16 f32): lanes 0-15 -> N=lane, K=0/1; lanes 16-31 -> K=2/3.
// C layout (16x16 f32, 8 VGPRs): lanes 0-15 rows 0..7, lanes 16-31 rows 8..15.
// ---------------------------------------------------------------------------
__global__ void gat_gemm_wmma(const float* __restrict__ X,
                              const float* __restrict__ W,
                              float* __restrict__ H, int K) {
    __shared__ float xs[16 * 128];
    const int row0 = blockIdx.x * 16;
    // cooperative stage of the contiguous 16-row x-tile
    for (int i = threadIdx.x; i < 16 * K; i += 256)
        xs[i] = X[(size_t)row0 * K + i];
    __syncthreads();

    const int wave = threadIdx.x >> 5;
    const int lane = threadIdx.x & 31;
    const int col0 = wave * 16;
    const int r    = lane & 15;          // M for A, N for B
    const int kb   = (lane >> 4) << 1;   // 0 or 2

    const float* __restrict__ xrow = xs + r * K;
    const float* __restrict__ wrow = W + (size_t)(col0 + r) * K;

    v8f acc = {};
    for (int k0 = 0; k0 < K; k0 += 4) {
        v2f a, b;
        a.x = xrow[k0 + kb];
        a.y = xrow[k0 + kb + 1];
        b.x = wrow[k0 + kb];
        b.y = wrow[k0 + kb + 1];
        acc = __builtin_amdgcn_wmma_f32_16x16x4_f32(
            false, a, false, b, (short)0, acc, false, false);
    }

    const int n     = lane & 15;
    const int mbase = (lane >> 4) * 8;
#pragma unroll
    for (int i = 0; i < 8; ++i)
        H[(size_t)(row0 + mbase + i) * HC + col0 + n] = acc[i];
}

// ---------------------------------------------------------------------------
// Per-node attention logit halves: one wave32 per (node, head).
//   a_dst[n,h] = <h[n,h,:], att[h,0:64]>   (x_i part)
//   a_src[n,h] = <h[n,h,:], att[h,64:128]> (x_j part)
// ---------------------------------------------------------------------------
__global__ void gat_node_alpha(const float* __restrict__ H,
                               const float* __restrict__ att,
                               float* __restrict__ a_dst,
                               float* __restrict__ a_src, int N) {
    const int gw = blockIdx.x * 8 + (threadIdx.x >> 5);
    if (gw >= N * NHEADS) return;
    const int lane = threadIdx.x & 31;
    const int node = gw >> 1, head = gw & 1;
    const float* hp = H + (size_t)node * HC + head * CC;
    const float* ad = att + head * (2 * CC);
    const float* as = ad + CC;
    const float h0 = hp[lane], h1 = hp[lane + 32];
    float pd = h0 * ad[lane] + h1 * ad[lane + 32];
    float ps = h0 * as[lane] + h1 * as[lane + 32];
#pragma unroll
    for (int off = 16; off; off >>= 1) {
        pd += __shfl_xor(pd, off, 32);
        ps += __shfl_xor(ps, off, 32);
    }
    if (lane == 0) { a_dst[gw] = pd; a_src[gw] = ps; }
}

// order-preserving float<->uint for atomicMax on floats
__device__ __forceinline__ unsigned f2ord(float f) {
    unsigned u = __float_as_uint(f);
    return (u & 0x80000000u) ? ~u : (u | 0x80000000u);
}
__device__ __forceinline__ float ord2f(unsigned u) {
    return __uint_as_float((u & 0x80000000u) ? (u & 0x7FFFFFFFu) : ~u);
}

__global__ void gat_init_scalars(unsigned* gmax, float* sums) {
    *gmax   = f2ord(-__builtin_inff());
    sums[0] = 0.f;
    sums[1] = 0.f;
}

// ---------------------------------------------------------------------------
// Pass 1 over edges (incl. appended self-loops): leaky-relu logits + global max
// ---------------------------------------------------------------------------
__global__ void gat_edge_alpha(const long long* __restrict__ ei, int E, int N,
                               const float* __restrict__ a_dst,
                               const float* __restrict__ a_src,
                               float* __restrict__ alpha,
                               unsigned* __restrict__ gmax) {
    const int e = blockIdx.x * blockDim.x + threadIdx.x;
    float m = -__builtin_inff();
    if (e < E + N) {
        int s, d;
        if (e < E) { s = (int)ei[e]; d = (int)ei[(size_t)E + e]; }
        else       { s = d = e - E; }
#pragma unroll
        for (int h = 0; h < NHEADS; ++h) {
            float a = a_dst[d * 2 + h] + a_src[s * 2 + h];
            a = (a >= 0.f) ? a : 0.2f * a;          // leaky_relu(0.2)
            alpha[(size_t)e * 2 + h] = a;
            m = fmaxf(m, a);
        }
    }
#pragma unroll
    for (int off = 16; off; off >>= 1) m = fmaxf(m, __shfl_xor(m, off, 32));
    if ((threadIdx.x & 31) == 0) atomicMax(gmax, f2ord(m));
}

// ---------------------------------------------------------------------------
// Pass 2: alpha = exp(alpha - max); per-head global sums
// ---------------------------------------------------------------------------
__global__ void gat_edge_expsum(float* __restrict__ alpha, int Etot,
                                const unsigned* __restrict__ gmax,
                                float* __restrict__ sums) {
    const int e = blockIdx.x * blockDim.x + threadIdx.x;
    float s0 = 0.f, s1 = 0.f;
    if (e < Etot) {
        const float mx = ord2f(*gmax);
        float a0 = __expf(alpha[(size_t)e * 2 + 0] - mx);
        float a1 = __expf(alpha[(size_t)e * 2 + 1] - mx);
        alpha[(size_t)e * 2 + 0] = a0;
        alpha[(size_t)e * 2 + 1] = a1;
        s0 = a0; s1 = a1;
    }
#pragma unroll
    for (int off = 16; off; off >>= 1) {
        s0 += __shfl_xor(s0, off, 32);
        s1 += __shfl_xor(s1, off, 32);
    }
    if ((threadIdx.x & 31) == 0) {
        atomicAdd(&sums[0], s0);
        atomicAdd(&sums[1], s1);
    }
}

// ---------------------------------------------------------------------------
// Weighted message scatter: aggr[dst,h,c] += h[src,h,c] * alpha/(sum+eps)
// 128 threads per edge (one per channel of H*C), 2 edges per block.
// ---------------------------------------------------------------------------
__global__ void gat_scatter(const long long* __restrict__ ei, int E, int N,
                            const float* __restrict__ H,
                            const float* __restrict__ alpha,
                            const float* __restrict__ sums,
                            float* __restrict__ aggr) {
    const int t = threadIdx.x & 127;
    const int e = blockIdx.x * 2 + (threadIdx.x >> 7);
    if (e >= E + N) return;
    int s, d;
    if (e < E) { s = (int)ei[e]; d = (int)ei[(size_t)E + e]; }
    else       { s = d = e - E; }
    const int head = t >> 6;
    const float w = alpha[(size_t)e * 2 + head] / (sums[head] + 1e-10f);
    atomicAdd(&aggr[(size_t)d * HC + t], H[(size_t)s * HC + t] * w);
}

// ---------------------------------------------------------------------------
// Head-mean + bias (+ optional relu)
// ---------------------------------------------------------------------------
__global__ void gat_finalize(const float* __restrict__ aggr,
                             const float* __restrict__ bias,
                             float* __restrict__ out, int N, int do_relu) {
    const int i = blockIdx.x * blockDim.x + threadIdx.x;
    if (i >= N * CC) return;
    const int n = i >> 6, c = i & 63;
    float v = 0.5f * (aggr[(size_t)n * HC + c] + aggr[(size_t)n * HC + CC + c])
              + bias[c];
    if (do_relu) v = fmaxf(v, 0.f);
    out[i] = v;
}

extern "C" void kernel_launch(void* const* d_in, const int* in_sizes, int n_in,
                              void* d_out, int out_size, void* d_ws, size_t ws_size,
                              hipStream_t stream) {
    const float*     x    = (const float*)d_in[0];
    const long long* ei   = (const long long*)d_in[1];   // int64 [2,E]
    const float*     W1   = (const float*)d_in[2];
    const float*     att1 = (const float*)d_in[3];
    const float*     b1   = (const float*)d_in[4];
    const float*     W2   = (const float*)d_in[5];
    const float*     att2 = (const float*)d_in[6];
    const float*     b2   = (const float*)d_in[7];
    float*           out  = (float*)d_out;

    const int N    = in_sizes[0] / HC;   // Fin == 128 == H*C
    const int E    = in_sizes[1] / 2;
    const int Etot = E + N;

    // workspace carve-up (floats)
    float* ws = (float*)d_ws;
    size_t off = 0;
    float* hbuf  = ws + off; off += (size_t)N * HC;      // GEMM output (per layer)
    float* aggr  = ws + off; off += (size_t)N * HC;      // scatter accumulator
    float* hmid  = ws + off; off += (size_t)N * CC;      // layer-1 output
    float* alpha = ws + off; off += (size_t)Etot * 2;    // per-edge, per-head
    float* adst  = ws + off; off += (size_t)N * 2;
    float* asrc  = ws + off; off += (size_t)N * 2;
    unsigned* gmax = (unsigned*)(ws + off); off += 1;
    float* sums  = ws + off; off += 2;

    for (int layer = 0; layer < 2; ++layer) {
        const float* xin  = layer ? hmid : x;
        const float* Wl   = layer ? W2 : W1;
        const float* attl = layer ? att2 : att1;
        const float* bl   = layer ? b2 : b1;
        const int    K    = layer ? CC : HC;             // 64 or 128
        float*       outl = layer ? out : hmid;

        hipMemsetAsync(aggr, 0, (size_t)N * HC * sizeof(float), stream);
        gat_init_scalars<<<1, 1, 0, stream>>>(gmax, sums);

        gat_gemm_wmma<<<N / 16, 256, 0, stream>>>(xin, Wl, hbuf, K);
        gat_node_alpha<<<(N * NHEADS + 7) / 8, 256, 0, stream>>>(hbuf, attl,
                                                                 adst, asrc, N);
        gat_edge_alpha<<<(Etot + 255) / 256, 256, 0, stream>>>(ei, E, N, adst,
                                                               asrc, alpha, gmax);
        gat_edge_expsum<<<(Etot + 255) / 256, 256, 0, stream>>>(alpha, Etot,
                                                                gmax, sums);
        gat_scatter<<<(Etot + 1) / 2, 256, 0, stream>>>(ei, E, N, hbuf, alpha,
                                                        sums, aggr);
        gat_finalize<<<(N * CC + 255) / 256, 256, 0, stream>>>(aggr, bl, outl,
                                                               N, layer == 0);
    }
}